// ProdLayer_69750268887705
// MI455X (gfx1250) — compile-verified
//
#include <hip/hip_runtime.h>
#include <stdint.h>

#define BATCH    512
#define N_PROD   65536
#define MAX_ELS  65600
#define N_EDGES  4

#define WAVES_PER_BLOCK 4              // 128 threads / block (4 wave32)
#define ROWS_PER_WAVE   8
#define THREADS         (WAVES_PER_BLOCK * 32)
#define COL_CHUNKS      (BATCH / 128)  // 4: 32 lanes x float4 = 128 floats/chunk

// Native vector types (HIP's float4 is a struct and is rejected by builtins).
typedef float v4f __attribute__((ext_vector_type(4)));
typedef int   v4i __attribute__((vector_size(4 * sizeof(int))));

// Address-space-qualified pointer element types for the async-to-LDS builtin:
// signature (from clang diagnostic): (v4i AS1*, v4i AS3*, imm offset, imm cpol).
typedef __attribute__((address_space(1))) v4i g_v4i;
typedef __attribute__((address_space(3))) v4i l_v4i;

// Issue one async global->LDS copy of 16 bytes per lane (512 B per wave).
__device__ __forceinline__ void async_load_b128(const float* gsrc, float* ldst) {
#if __has_builtin(__builtin_amdgcn_global_load_async_to_lds_b128)
    __builtin_amdgcn_global_load_async_to_lds_b128(
        (g_v4i*)gsrc, (l_v4i*)ldst, /*offset=*/0, /*cpol=*/0);
#else
    unsigned lds_off = (unsigned)(uintptr_t)(l_v4i*)ldst;
    asm volatile("global_load_async_to_lds_b128 %0, %1, off"
                 :: "v"(lds_off), "v"(gsrc)
                 : "memory");
#endif
}

#define WAIT_ASYNC_ALL()  asm volatile("s_wait_asynccnt 0"  ::: "memory")
#define WAIT_ASYNC_16()   asm volatile("s_wait_asynccnt 16" ::: "memory")

__global__ __launch_bounds__(THREADS) void prod_forward_async(
    const float* __restrict__ node_mars,
    const long long* __restrict__ nids,
    const long long* __restrict__ cids,
    float* __restrict__ out)
{
    // Per-wave double buffer: 2 x (4 child rows x 512 floats) = 16 KB/wave.
    __shared__ __align__(16) float lds[WAVES_PER_BLOCK][2][N_EDGES * BATCH];

    const int lane = threadIdx.x & 31;
    const int wave = __builtin_amdgcn_readfirstlane((int)(threadIdx.x >> 5));
    const int baseRow = (blockIdx.x * WAVES_PER_BLOCK + wave) * ROWS_PER_WAVE;

    float* buf0 = &lds[wave][0][0];
    float* buf1 = &lds[wave][1][0];

    auto issue = [&](int row, float* buf) {
        #pragma unroll
        for (int e = 0; e < N_EDGES; ++e) {
            // Uniform child index -> SGPR (saddr + voffset addressing).
            int ci = __builtin_amdgcn_readfirstlane(
                (int)cids[(size_t)row * N_EDGES + e]);
            const float* src = node_mars + (size_t)ci * BATCH + lane * 4;
            float*       dst = buf + e * BATCH + lane * 4;
            #pragma unroll
            for (int k = 0; k < COL_CHUNKS; ++k)
                async_load_b128(src + k * 128, dst + k * 128);
        }
    };

    // Prime the pipeline with row 0 of this wave.
    issue(baseRow, buf0);

    for (int r = 0; r < ROWS_PER_WAVE; ++r) {
        float* cur = (r & 1) ? buf1 : buf0;
        float* nxt = (r & 1) ? buf0 : buf1;

        if (r + 1 < ROWS_PER_WAVE) {
            issue(baseRow + r + 1, nxt);   // 16 new async ops in flight
            WAIT_ASYNC_16();               // in-order: current row's 16 are done
        } else {
            WAIT_ASYNC_ALL();
        }

        const int outRow = __builtin_amdgcn_readfirstlane(
            (int)nids[baseRow + r]);
        float* dstRow = out + (size_t)outRow * BATCH;

        #pragma unroll
        for (int k = 0; k < COL_CHUNKS; ++k) {
            const int col = k * 128 + lane * 4;
            v4f a = *(const v4f*)(cur + 0 * BATCH + col);
            v4f b = *(const v4f*)(cur + 1 * BATCH + col);
            v4f c = *(const v4f*)(cur + 2 * BATCH + col);
            v4f d = *(const v4f*)(cur + 3 * BATCH + col);
            v4f s = (a + b) + (c + d);
            // Output is streamed once and never re-read: NT-hint it so the
            // 192 MB L2 stays available for node_mars gather reuse.
            __builtin_nontemporal_store(s, (v4f*)(dstRow + col));
        }
    }
}

// Rows [N_PROD, MAX_ELS) are untouched by the scatter (nids == arange(N_PROD)):
// copy them from the input element_mars (d_out is poisoned by the harness).
__global__ void tail_copy(const float* __restrict__ element_mars,
                          float* __restrict__ out)
{
    const int n4 = (MAX_ELS - N_PROD) * BATCH / 4;   // 8192 float4s
    int i = blockIdx.x * blockDim.x + threadIdx.x;
    if (i < n4) {
        size_t base = (size_t)N_PROD * BATCH;
        v4f v = ((const v4f*)(element_mars + base))[i];
        __builtin_nontemporal_store(v, (v4f*)(out + base) + i);
    }
}

extern "C" void kernel_launch(void* const* d_in, const int* in_sizes, int n_in,
                              void* d_out, int out_size, void* d_ws, size_t ws_size,
                              hipStream_t stream) {
    const float*     node_mars    = (const float*)d_in[0];
    const float*     element_mars = (const float*)d_in[1];
    const long long* nids         = (const long long*)d_in[2];
    const long long* cids         = (const long long*)d_in[3];
    float*           out          = (float*)d_out;

    const int nblocks = N_PROD / (WAVES_PER_BLOCK * ROWS_PER_WAVE); // 2048
    prod_forward_async<<<nblocks, THREADS, 0, stream>>>(node_mars, nids, cids, out);

    const int n4 = (MAX_ELS - N_PROD) * BATCH / 4;
    tail_copy<<<(n4 + 127) / 128, 128, 0, stream>>>(element_mars, out);
}